// Decoder_20624432956209
// MI455X (gfx1250) — compile-verified
//
#include <hip/hip_runtime.h>
#include <stdint.h>

// Greedy flat-NMS span decoder for MI455X (gfx1250, wave32).
// One workgroup per batch row; everything lives in LDS (320KB/WGP).
// probs row is staged into LDS via the Tensor Data Mover (TENSORcnt path).

#define B_      4
#define NSPAN   1024
#define NENT    8
#define NCAND   (NSPAN * NENT)     // 8192
#define NTHR    1024               // 32 waves of 32
#define EPT     (NCAND / NTHR)     // 8 candidates per thread
#define THRESH  0.5f

typedef uint32_t u32;
typedef uint64_t u64;

typedef __attribute__((ext_vector_type(4))) u32 v4u;
typedef __attribute__((ext_vector_type(8))) int v8i;
typedef __attribute__((ext_vector_type(4))) int v4i;

// Dynamic LDS layout (bytes):
//   [0      , 32768 ) f32 sprob[8192]  -- TDM staging, later reused as output accumulator
//   [32768  , 98304 ) u64 skey[8192]   -- sort keys
//   [98304  , 102400) i32 s_sp[1024]   -- span starts
//   [102400 , 106496) i32 e_sp[1024]   -- span ends
//   [106496 , 114688) u8  alive[8192]
//   [114688 , 114692) u32 minPos
#define LDS_BYTES 114944

__global__ __launch_bounds__(NTHR)
void nms_decode_kernel(const float* __restrict__ probs,
                       const int* __restrict__ spans,
                       float* __restrict__ out)
{
    extern __shared__ char smem[];
    float* sprob = (float*)smem;
    float* soutv = (float*)smem;                 // alias: probs dead after key build
    u64*   skey  = (u64*)(smem + 32768);
    int*   s_sp  = (int*)(smem + 98304);
    int*   e_sp  = (int*)(smem + 102400);
    unsigned char* alive = (unsigned char*)(smem + 106496);
    u32*   minPos = (u32*)(smem + 114688);

    const int tid = threadIdx.x;
    const int b   = blockIdx.x;
    const float* gp = probs + (size_t)b * NCAND;

    // ---- Stage probs[b] (32 KB contiguous) into LDS with the Tensor Data Mover ----
#if __has_builtin(__builtin_amdgcn_tensor_load_to_lds) && __has_builtin(__builtin_amdgcn_s_wait_tensorcnt)
    if (tid < 32) {  // whole wave 0, EXEC all-ones inside this branch
        u64 ga  = (u64)(uintptr_t)gp;
        u32 lds = (u32)(uintptr_t)(void*)sprob;  // low 32 bits of flat ptr = LDS byte offset
        // D# group 0: count=1, lds_addr, global_addr[56:0], type=2 ("image")
        v4u g0 = { 1u,
                   lds,
                   (u32)ga,
                   ((u32)(ga >> 32) & 0x01FFFFFFu) | (2u << 30) };
        // D# group 1: data_size=4B (code 2), 1-D tile: tensor_dim0=8192,
        // tensor_dim1=1, tile_dim0=8192, tile_dim1/2=0, dim0_stride=8192.
        v8i g1 = { (int)(2u << 16),      // wg_mask=0 | data_size=2 (4B)
                   (int)0x20000000u,     // tensor_dim0 low16 (0x2000) in [31:16]
                   (int)0x00010000u,     // tensor_dim0 hi16=0 | tensor_dim1 low16=1
                   (int)0x20000000u,     // tensor_dim1 hi16=0 | tile_dim0=0x2000
                   0,                    // tile_dim1=0, tile_dim2=0
                   8192,                 // tensor_dim0_stride[31:0]
                   0, 0 };
        v4i gz4 = { 0, 0, 0, 0 };
        v8i gz8 = { 0, 0, 0, 0, 0, 0, 0, 0 };
        // 6-arg form (clang-23 / therock-10.0 toolchain)
        __builtin_amdgcn_tensor_load_to_lds(g0, g1, gz4, gz4, gz8, 0);
        __builtin_amdgcn_s_wait_tensorcnt(0);
    }
    __syncthreads();
#define PROB_AT(i) sprob[i]
#else
#define PROB_AT(i) gp[i]
#endif

    // ---- Spans into LDS (one span per thread) ----
    {
        const int* sp = spans + (size_t)b * NSPAN * 2;
        s_sp[tid] = sp[2 * tid];
        e_sp[tid] = sp[2 * tid + 1];
    }

    // ---- Build sort keys; init alive flags; zero output accumulator ----
    // key = (valid ? ~float_bits(score) : 0xFFFFFFFF) << 32 | candidate_index
    // Ascending sort of key == descending score, stable by index; invalid sink
    // to the tail (their output is 0 regardless of internal order).
    for (int m = 0; m < EPT; ++m) {
        int g = tid + m * NTHR;
        float p = PROB_AT(g);
        bool valid = p > THRESH;
        u32 hi = valid ? ~__float_as_uint(p) : 0xFFFFFFFFu;
        skey[g]  = ((u64)hi << 32) | (u32)g;
        alive[g] = valid ? 1 : 0;
        soutv[g] = 0.0f;   // same address as sprob[g]; same thread, after the read
    }

    // ---- Bitonic sort of 8192 64-bit keys in LDS (ascending) ----
    for (u32 k = 2; k <= NCAND; k <<= 1) {
        for (u32 j = k >> 1; j > 0; j >>= 1) {
            __syncthreads();
            for (int m = 0; m < EPT; ++m) {
                u32 i = (u32)tid + (u32)(m * NTHR);
                u32 l = i ^ j;
                if (l > i) {
                    u64 a = skey[i];
                    u64 c = skey[l];
                    bool up = ((i & k) == 0);
                    if ((a > c) == up) { skey[i] = c; skey[l] = a; }
                }
            }
        }
    }
    __syncthreads();

    // ---- Round-parallel greedy suppression (exact for 'flat' mode):
    // the lowest sorted position still alive is the next kept span.
    const int lane = tid & 31;
    for (int round = 0; round < NCAND; ++round) {
        if (tid == 0) *minPos = 0xFFFFFFFFu;
        __syncthreads();

        u32 lm = 0xFFFFFFFFu;
        for (int m = 0; m < EPT; ++m) {           // positions ascend with m
            u32 q = (u32)tid + (u32)(m * NTHR);
            if (alive[q]) { lm = q; break; }
        }
        // wave32 min-reduce, then one LDS atomic per wave
        for (int off = 16; off > 0; off >>= 1) {
            u32 o = (u32)__shfl_xor((int)lm, off, 32);
            lm = lm < o ? lm : o;
        }
        if (lane == 0) atomicMin(minPos, lm);
        __syncthreads();

        u32 p = *minPos;
        if (p == 0xFFFFFFFFu) break;              // uniform exit

        u64 key = skey[p];
        u32 idx = (u32)key;
        int span = (int)(idx >> 3);
        int s0 = s_sp[span], e0 = e_sp[span];
        if (tid == 0) soutv[p] = __uint_as_float(~(u32)(key >> 32));

        // suppress everything overlapping the kept span (kills p itself too)
        for (int m = 0; m < EPT; ++m) {
            u32 q = (u32)tid + (u32)(m * NTHR);
            if (alive[q]) {
                int sp2 = (int)(((u32)skey[q]) >> 3);
                int s1 = s_sp[sp2], e1 = e_sp[sp2];
                if (!((e0 < s1) || (e1 < s0))) alive[q] = 0;
            }
        }
        __syncthreads();   // all reads of *minPos / writes of alive done before reset
    }
    __syncthreads();

    // ---- Write result row (coalesced) ----
    float* go = out + (size_t)b * NCAND;
    for (int m = 0; m < EPT; ++m) {
        int g = tid + m * NTHR;
        go[g] = soutv[g];
    }
}

extern "C" void kernel_launch(void* const* d_in, const int* in_sizes, int n_in,
                              void* d_out, int out_size, void* d_ws, size_t ws_size,
                              hipStream_t stream) {
    (void)in_sizes; (void)n_in; (void)out_size; (void)d_ws; (void)ws_size;
    const float* probs = (const float*)d_in[0];
    const int*   spans = (const int*)d_in[1];
    float*       out   = (float*)d_out;
    hipLaunchKernelGGL(nms_decode_kernel, dim3(B_), dim3(NTHR), LDS_BYTES, stream,
                       probs, spans, out);
}